// yoloLoss_65335042507299
// MI455X (gfx1250) — compile-verified
//
#include <hip/hip_runtime.h>
#include <hip/hip_bf16.h>

#ifndef __has_builtin
#define __has_builtin(x) 0
#endif

#if __has_builtin(__builtin_amdgcn_global_load_async_to_lds_b128)
#define USE_ASYNC_LDS 1
#else
#define USE_ASYNC_LDS 0
#endif

#define S_GRID 14
#define CELL_F 30                   // floats per cell per tensor
#define CPB 256                     // cells per block (== threads per block)
#define BLK_F (CPB * CELL_F)        // 7680 floats per tensor per block

// Exact parameter type expected by the async-LDS builtin per hipcc diagnostic:
// '__attribute__((__vector_size__(4 * sizeof(int)))) int *'
typedef int v4i __attribute__((vector_size(16)));

__device__ __forceinline__ void stage16(const float* g, float* l) {
#if USE_ASYNC_LDS
    // gfx1250 async copy: 16B global -> LDS, no VGPR round trip, ASYNCcnt-tracked
    __builtin_amdgcn_global_load_async_to_lds_b128((v4i*)g, (v4i*)l, 0, 0);
#else
    *(float4*)l = *(const float4*)g;
#endif
}

__device__ __forceinline__ void stage_wait() {
#if USE_ASYNC_LDS
#if __has_builtin(__builtin_amdgcn_s_wait_asynccnt)
    __builtin_amdgcn_s_wait_asynccnt(0);
#else
    asm volatile("s_wait_asynccnt 0" ::: "memory");
#endif
#endif
}

// Faithful translation of the per-cell loss from the reference.
__device__ float cell_loss(const float* __restrict__ p,
                           const float* __restrict__ t,
                           int gi, int gj) {
    const float STEP = 1.0f / (float)S_GRID;
    const float EPS  = 1e-6f;
    const float FOUR_OVER_PI2 = 0.405284734569351086f;   // 4/pi^2

    float iou[2], ciouL[2], dconf[2];
    bool  obj[2];

#pragma unroll
    for (int b = 0; b < 2; ++b) {
        const float px = p[b * 5 + 0], py = p[b * 5 + 1];
        const float pw = p[b * 5 + 2], ph = p[b * 5 + 3];
        const float pc = p[b * 5 + 4];
        const float tx = t[b * 5 + 0], ty = t[b * 5 + 1];
        const float tw = t[b * 5 + 2], th = t[b * 5 + 3];
        const float tc = t[b * 5 + 4];

        // xywh -> xyxy (gx = column index j, gy = row index i)
        const float pcx = (px + (float)gj) * STEP, pcy = (py + (float)gi) * STEP;
        const float tcx = (tx + (float)gj) * STEP, tcy = (ty + (float)gi) * STEP;
        const float b1x1 = pcx - 0.5f * pw, b1y1 = pcy - 0.5f * ph;
        const float b1x2 = pcx + 0.5f * pw, b1y2 = pcy + 0.5f * ph;
        const float b2x1 = tcx - 0.5f * tw, b2y1 = tcy - 0.5f * th;
        const float b2x2 = tcx + 0.5f * tw, b2y2 = tcy + 0.5f * th;

        // IoU
        const float ix1 = fmaxf(b1x1, b2x1), iy1 = fmaxf(b1y1, b2y1);
        const float ix2 = fminf(b1x2, b2x2), iy2 = fminf(b1y2, b2y2);
        const float inter = fmaxf(ix2 - ix1, 0.0f) * fmaxf(iy2 - iy1, 0.0f);
        const float a1 = (b1x2 - b1x1) * (b1y2 - b1y1);
        const float a2 = (b2x2 - b2x1) * (b2y2 - b2y1);
        const float un = a1 + a2 - inter + EPS;
        const float io = inter / un;
        iou[b] = io;

        // CIoU loss
        const float cx1 = 0.5f * (b1x1 + b1x2), cy1 = 0.5f * (b1y1 + b1y2);
        const float cx2 = 0.5f * (b2x1 + b2x2), cy2 = 0.5f * (b2y1 + b2y2);
        const float dcx = cx1 - cx2, dcy = cy1 - cy2;
        const float center_dist = dcx * dcx + dcy * dcy;
        const float ox1 = fminf(b1x1, b2x1), oy1 = fminf(b1y1, b2y1);
        const float ox2 = fmaxf(b1x2, b2x2), oy2 = fmaxf(b1y2, b2y2);
        const float odx = ox2 - ox1, ody = oy2 - oy1;
        const float outer_diag = odx * odx + ody * ody + EPS;
        const float w1 = fmaxf(b1x2 - b1x1, EPS), h1 = fmaxf(b1y2 - b1y1, EPS);
        const float w2 = fmaxf(b2x2 - b2x1, EPS), h2 = fmaxf(b2y2 - b2y1, EPS);
        const float dat = atanf(w2 / h2) - atanf(w1 / h1);
        const float v = FOUR_OVER_PI2 * dat * dat;
        const float alpha = v / (1.0f - io + v + EPS);
        const float ciou = io - center_dist / outer_diag - alpha * v;
        const float scale = fmaxf(2.0f - w2 * h2, 1.0f);
        ciouL[b] = (1.0f - ciou) * scale;

        const float dc = pc - tc;
        dconf[b] = dc * dc;
        obj[b] = (tc > 0.0f);
    }

    const bool sig = obj[1];
    const int resp = (iou[1] > iou[0]) ? 1 : 0;   // argmax, tie -> 0

    float loss = 0.0f;
#pragma unroll
    for (int b = 0; b < 2; ++b) {
        const bool suppress = sig && (b == (1 - resp));
        const bool obj_m = obj[b] && !suppress;
        loss += obj_m ? dconf[b] : 0.2f * dconf[b];       // obj + LAMBDA_NOOBJ*noobj
        if (sig && obj_m) loss += 7.0f * ciouL[b];        // LAMBDA_COORD*ciou
    }

    if (sig) {
        float bce = 0.0f;
#pragma unroll
        for (int k = 0; k < 20; ++k) {
            const float tc = t[10 + k];
            const float pc = fminf(fmaxf(p[10 + k], 1e-7f), 1.0f - 1e-7f);
            // __logf -> single v_log_f32 (TRANS32, co-executes with VALU on CDNA5).
            // p is clipped to [1e-7, 1-1e-7], so log(1-p) == log1p(-p) to f32 accuracy.
            bce -= tc * __logf(pc) + (1.0f - tc) * __logf(1.0f - pc);
        }
        loss += 1.5f * bce;                               // LAMBDA_CLS
    }
    return loss;
}

__global__ __launch_bounds__(CPB) void yolo_main(const float* __restrict__ pred,
                                                 const float* __restrict__ targ,
                                                 float* __restrict__ partials,
                                                 int n_cells) {
    __shared__ __align__(16) float sP[BLK_F];
    __shared__ __align__(16) float sT[BLK_F];
    __shared__ float sRed[CPB / 32];

    const int tid = threadIdx.x;
    const int blockCell = blockIdx.x * CPB;
    const float* gP = pred + (size_t)blockCell * CELL_F;
    const float* gT = targ + (size_t)blockCell * CELL_F;

    const int remCells = min(CPB, n_cells - blockCell);
    const int remF4 = (remCells * CELL_F) / 4;            // CELL_F*CPB % 4 == 0 per block

    // Stage this block's 60KB slice into LDS with coalesced b128 async transfers.
    for (int t4 = tid; t4 < remF4; t4 += CPB) {
        stage16(gP + t4 * 4, sP + t4 * 4);
        stage16(gT + t4 * 4, sT + t4 * 4);
    }
    stage_wait();
    __syncthreads();

    float contrib = 0.0f;
    const int c = blockCell + tid;
    if (tid < remCells) {
        const int ij = c % (S_GRID * S_GRID);
        const int gj = ij % S_GRID;
        const int gi = ij / S_GRID;
        contrib = cell_loss(sP + tid * CELL_F, sT + tid * CELL_F, gi, gj);
    }

    // wave32 reduction
#pragma unroll
    for (int off = 16; off > 0; off >>= 1)
        contrib += __shfl_xor(contrib, off, 32);
    if ((tid & 31) == 0) sRed[tid >> 5] = contrib;
    __syncthreads();
    if (tid == 0) {
        float s = 0.0f;
#pragma unroll
        for (int w = 0; w < CPB / 32; ++w) s += sRed[w];
        partials[blockIdx.x] = s;
    }
}

__global__ __launch_bounds__(256) void yolo_reduce(const float* __restrict__ partials,
                                                   int n, float inv_n_images,
                                                   float* __restrict__ out) {
    __shared__ float sRed[8];
    float acc = 0.0f;
    for (int k = threadIdx.x; k < n; k += 256) acc += partials[k];
#pragma unroll
    for (int off = 16; off > 0; off >>= 1)
        acc += __shfl_xor(acc, off, 32);
    if ((threadIdx.x & 31) == 0) sRed[threadIdx.x >> 5] = acc;
    __syncthreads();
    if (threadIdx.x == 0) {
        float s = 0.0f;
#pragma unroll
        for (int w = 0; w < 8; ++w) s += sRed[w];
        out[0] = s * inv_n_images;
    }
}

extern "C" void kernel_launch(void* const* d_in, const int* in_sizes, int n_in,
                              void* d_out, int out_size, void* d_ws, size_t ws_size,
                              hipStream_t stream) {
    (void)n_in; (void)out_size; (void)ws_size;
    const float* pred = (const float*)d_in[0];
    const float* targ = (const float*)d_in[1];
    float* out = (float*)d_out;
    float* partials = (float*)d_ws;

    const int n_cells = in_sizes[0] / CELL_F;          // 4096*14*14 = 802816
    const int n_images = n_cells / (S_GRID * S_GRID);  // 4096
    const int blocks = (n_cells + CPB - 1) / CPB;      // 3136

    yolo_main<<<blocks, CPB, 0, stream>>>(pred, targ, partials, n_cells);
    yolo_reduce<<<1, 256, 0, stream>>>(partials, blocks, 1.0f / (float)n_images, out);
}